// WrappedMultiConv_5111011082637
// MI455X (gfx1250) — compile-verified
//
#include <hip/hip_runtime.h>

typedef __attribute__((ext_vector_type(2))) float v2f;
typedef __attribute__((ext_vector_type(4))) float v4f;
typedef __attribute__((ext_vector_type(8))) float v8f;

#define ASTRIDE 36   // padded A row stride (floats): bank-conflict free fragment reads

// ---------------------------------------------------------------------------
// Utility kernels
// ---------------------------------------------------------------------------
__global__ void k_zero(float* __restrict__ p, int n) {
    int i = blockIdx.x * blockDim.x + threadIdx.x;
    if (i < n) p[i] = 0.0f;
}

// out[n][c] = bias0[c] + bias1[c] + bias2[c]  (fold all 3 conv biases)
__global__ void k_init_out(float* __restrict__ out, const float* __restrict__ bias, int n128) {
    int i = blockIdx.x * blockDim.x + threadIdx.x;
    if (i < n128) {
        int c = i & 127;
        out[i] = bias[c] + bias[128 + c] + bias[256 + c];
    }
}

// deg[src[e]] += w[e]
__global__ void k_deg(const long long* __restrict__ src, const float* __restrict__ w,
                      float* __restrict__ deg, int E) {
    int e = blockIdx.x * blockDim.x + threadIdx.x;
    if (e < E) unsafeAtomicAdd(&deg[(int)src[e]], w[e]);
}

// y[dst[e]] += -w[e] * v[src[e]]   (one wave of 32 lanes per edge, v4f/lane)
__global__ void k_edge_scatter(const float* __restrict__ v, float* __restrict__ y,
                               const long long* __restrict__ src,
                               const long long* __restrict__ dst,
                               const float* __restrict__ w, int E) {
    long long g = (long long)blockIdx.x * blockDim.x + threadIdx.x;
    int e = (int)(g >> 5);
    if (e >= E) return;
    int lane = (int)(g & 31);
    int s = (int)src[e];
    int d = (int)dst[e];
    float ww = -w[e];
    v4f xv = ((const v4f*)(v + (size_t)s * 128))[lane];
    float* yp = y + (size_t)d * 128 + lane * 4;
    unsafeAtomicAdd(yp + 0, ww * xv.x);
    unsafeAtomicAdd(yp + 1, ww * xv.y);
    unsafeAtomicAdd(yp + 2, ww * xv.z);
    unsafeAtomicAdd(yp + 3, ww * xv.w);
}

// tx1 += (deg-1) * x      (v4f granularity)
__global__ void k_diag_tx1(v4f* __restrict__ tx1, const v4f* __restrict__ x,
                           const float* __restrict__ deg, int n32) {
    int i = blockIdx.x * blockDim.x + threadIdx.x;
    if (i < n32) {
        float dgl = deg[i >> 5] - 1.0f;
        tx1[i] = tx1[i] + dgl * x[i];
    }
}

// tx2 = 2*(tx2_scatter + (deg-1)*tx1) - x
__global__ void k_fin_tx2(v4f* __restrict__ tx2, const v4f* __restrict__ tx1,
                          const v4f* __restrict__ x, const float* __restrict__ deg, int n32) {
    int i = blockIdx.x * blockDim.x + threadIdx.x;
    if (i < n32) {
        float dgl = deg[i >> 5] - 1.0f;
        tx2[i] = 2.0f * (tx2[i] + dgl * tx1[i]) - x[i];
    }
}

// ---------------------------------------------------------------------------
// Fused GEMM: out[128-row tile] += T0@W[0] + T1@W[1] + T2@W[2]
// [N,384] @ [384,128] with K concatenated over the 3 Chebyshev terms.
// 8 waves/block; each wave owns 16 rows x 128 cols = 8 WMMA accumulators.
// Inner k-steps are software-pipelined (next A + 8 B fragments loaded during
// the current 8 WMMAs); staging uses ext_vector types so selects stay in VGPRs.
// ---------------------------------------------------------------------------
__global__ __launch_bounds__(256, 1)
void k_gemm(const float* __restrict__ T0, const float* __restrict__ T1,
            const float* __restrict__ T2, const float* __restrict__ W,
            float* __restrict__ out, int N) {
    __shared__ float a_lds[128 * ASTRIDE];   // 18 KB: 128 rows x 32-K chunk (padded)
    __shared__ float w_lds[16 * 256];        // 16 KB: (k,k+1) pairs x 128 cols

    const int tid  = threadIdx.x;
    const int wave = tid >> 5;
    const int lane = tid & 31;
    const int lhi  = lane >> 4;    // 0: K pair {0,1}, 1: K pair {2,3}
    const int llo  = lane & 15;
    const int rowBase = blockIdx.x * 128;
    const int waveRow = wave * 16;

    const v8f zero = {0.f, 0.f, 0.f, 0.f, 0.f, 0.f, 0.f, 0.f};
    v8f acc[8];
#pragma unroll
    for (int j = 0; j < 8; ++j) acc[j] = zero;

    // per-thread staging coordinates
    const int sRow = tid >> 1;             // A: 2 threads / row
    const int sC0  = (tid & 1) * 16;
    const int sGr  = rowBase + sRow;
    const bool sOk = (sGr < N);
    // clamped safe row offset for OOB threads (data zeroed at LDS-store time)
    const size_t aRowOff = (size_t)(sOk ? sGr : 0) * 128 + sC0;
    const int wKl  = tid >> 3;             // W: local k 0..31
    const int wC0  = (tid & 7) * 16;

    // fragment base pointers
    const float* aFragBase = a_lds + (waveRow + llo) * ASTRIDE + 2 * lhi;
    const float* bFragBase = w_lds + lhi * 256 + llo * 2;

    for (int t = 0; t < 12; ++t) {         // 12 chunks = 3 K-blocks x 4 chunks of 32
        const int kb = t >> 2;
        const int cc = t & 3;
        const float* Ab = (kb == 0) ? T0 : ((kb == 1) ? T1 : T2);

        __syncthreads();                   // previous chunk's LDS consumers done
        // ---- stage chunk t: global -> short-lived regs -> LDS ----
        {
            const v4f z4 = {0.f, 0.f, 0.f, 0.f};
            const v4f* gpA = (const v4f*)(Ab + aRowOff + cc * 32);
            const v4f* gpW = (const v4f*)(W + (size_t)kb * 16384 +
                                          (size_t)(cc * 32 + wKl) * 128 + wC0);
            v4f aR0 = gpA[0];
            v4f aR1 = gpA[1];
            v4f aR2 = gpA[2];
            v4f aR3 = gpA[3];
            v4f wR0 = gpW[0];
            v4f wR1 = gpW[1];
            v4f wR2 = gpW[2];
            v4f wR3 = gpW[3];

            v4f* lp = (v4f*)(a_lds + sRow * ASTRIDE + sC0);
            lp[0] = sOk ? aR0 : z4;        // ext_vector select: stays in VGPRs
            lp[1] = sOk ? aR1 : z4;
            lp[2] = sOk ? aR2 : z4;
            lp[3] = sOk ? aR3 : z4;

            const int base = (wKl >> 1) * 256 + (wKl & 1) + wC0 * 2;
            w_lds[base +  0] = wR0.x;  w_lds[base +  2] = wR0.y;
            w_lds[base +  4] = wR0.z;  w_lds[base +  6] = wR0.w;
            w_lds[base +  8] = wR1.x;  w_lds[base + 10] = wR1.y;
            w_lds[base + 12] = wR1.z;  w_lds[base + 14] = wR1.w;
            w_lds[base + 16] = wR2.x;  w_lds[base + 18] = wR2.y;
            w_lds[base + 20] = wR2.z;  w_lds[base + 22] = wR2.w;
            w_lds[base + 24] = wR3.x;  w_lds[base + 26] = wR3.y;
            w_lds[base + 28] = wR3.z;  w_lds[base + 30] = wR3.w;
        }
        __syncthreads();
        // ---- prefetch chunk t+1 into cache (no registers held) ----
        if (t + 1 < 12) {
            const int kb2 = (t + 1) >> 2;
            const int cc2 = (t + 1) & 3;
            const float* Ab2 = (kb2 == 0) ? T0 : ((kb2 == 1) ? T1 : T2);
            __builtin_prefetch(Ab2 + aRowOff + cc2 * 32, 0, 3);
            __builtin_prefetch(W + (size_t)kb2 * 16384 + (size_t)(cc2 * 32 + wKl) * 128 + wC0, 0, 3);
        }
        // ---- compute: 8 K-steps of 4, fragments pipelined one step ahead ----
        v2f a_cur = *(const v2f*)(aFragBase + 0);
        v2f b_cur[8];
#pragma unroll
        for (int j = 0; j < 8; ++j)
            b_cur[j] = *(const v2f*)(bFragBase + j * 32);
#pragma unroll
        for (int kk = 0; kk < 32; kk += 4) {
            v2f a_nxt;
            v2f b_nxt[8];
            if (kk + 4 < 32) {
                a_nxt = *(const v2f*)(aFragBase + kk + 4);
#pragma unroll
                for (int j = 0; j < 8; ++j)
                    b_nxt[j] = *(const v2f*)(bFragBase + ((kk + 4) >> 1) * 256 + j * 32);
            }
            // A frag (ISA 16x4 f32): lane m<16 -> {A[m][k],A[m][k+1]},
            // lane m+16 -> {A[m][k+2],A[m][k+3]}
#pragma unroll
            for (int j = 0; j < 8; ++j)
                acc[j] = __builtin_amdgcn_wmma_f32_16x16x4_f32(
                    false, a_cur, false, b_cur[j], (short)0, acc[j], false, false);
            if (kk + 4 < 32) {
                a_cur = a_nxt;
#pragma unroll
                for (int j = 0; j < 8; ++j) b_cur[j] = b_nxt[j];
            }
        }
    }
    // ---- epilogue: C/D layout VGPR r = rows {r, r+8}, N = llo ----
#pragma unroll
    for (int j = 0; j < 8; ++j) {
#pragma unroll
        for (int r = 0; r < 8; ++r) {
            int row = rowBase + waveRow + r + lhi * 8;
            if (row < N) out[(size_t)row * 128 + j * 16 + llo] += acc[j][r];
        }
    }
}

// ---------------------------------------------------------------------------
extern "C" void kernel_launch(void* const* d_in, const int* in_sizes, int n_in,
                              void* d_out, int out_size, void* d_ws, size_t ws_size,
                              hipStream_t stream) {
    const float*     x    = (const float*)d_in[0];
    const long long* ei   = (const long long*)d_in[1];  // [NC, 2, E] int64
    const float*     ew   = (const float*)d_in[2];      // [NC, E]
    const float*     Wt   = (const float*)d_in[3];      // [NC, K=3, 128, 128]
    const float*     bias = (const float*)d_in[4];      // [NC, 128]
    float*           out  = (float*)d_out;

    const int Cin = 128;
    const int NC  = 3;
    const int N   = in_sizes[0] / Cin;
    const int E   = in_sizes[2] / NC;

    // workspace carve-out
    char*  ws    = (char*)d_ws;
    size_t t1Off = (((size_t)N * 4) + 255) & ~(size_t)255;
    size_t t2Off = t1Off + ((((size_t)N * Cin * 4) + 255) & ~(size_t)255);
    float* deg = (float*)(ws);
    float* tx1 = (float*)(ws + t1Off);
    float* tx2 = (float*)(ws + t2Off);

    const int TPB    = 256;
    const int n128   = N * 128;
    const int n32    = N * 32;
    const int bN     = (N + TPB - 1) / TPB;
    const int bE     = (E + TPB - 1) / TPB;
    const int bN128  = (n128 + TPB - 1) / TPB;
    const int bN32   = (n32 + TPB - 1) / TPB;
    const int bEdge  = (int)(((long long)E * 32 + TPB - 1) / TPB);
    const int bGemm  = (N + 127) / 128;

    k_init_out<<<bN128, TPB, 0, stream>>>(out, bias, n128);

    for (int i = 0; i < NC; ++i) {
        const long long* src = ei + (size_t)i * 2 * E;
        const long long* dst = src + E;
        const float*     w   = ew + (size_t)i * E;
        const float*     Wi  = Wt + (size_t)i * 3 * 128 * 128;

        // deg = segment_sum(w, src)
        k_zero<<<bN, TPB, 0, stream>>>(deg, N);
        k_deg<<<bE, TPB, 0, stream>>>(src, w, deg, E);

        // Tx1 = L_hat @ x
        k_zero<<<bN128, TPB, 0, stream>>>(tx1, n128);
        k_edge_scatter<<<bEdge, TPB, 0, stream>>>(x, tx1, src, dst, w, E);
        k_diag_tx1<<<bN32, TPB, 0, stream>>>((v4f*)tx1, (const v4f*)x, deg, n32);

        // Tx2 = 2*(L_hat @ Tx1) - x
        k_zero<<<bN128, TPB, 0, stream>>>(tx2, n128);
        k_edge_scatter<<<bEdge, TPB, 0, stream>>>(tx1, tx2, src, dst, w, E);
        k_fin_tx2<<<bN32, TPB, 0, stream>>>((v4f*)tx2, (const v4f*)tx1,
                                            (const v4f*)x, deg, n32);

        // out += x@W0 + Tx1@W1 + Tx2@W2
        k_gemm<<<bGemm, TPB, 0, stream>>>(x, tx1, tx2, Wi, out, N);
    }
}